// STMambaBlock_4904852652264
// MI455X (gfx1250) — compile-verified
//
#include <hip/hip_runtime.h>
#include <math.h>

// ---------------- problem constants ----------------
#define B_    2
#define T_    8
#define C_    256
#define H_    24
#define W_    24
#define L_    (T_*H_*W_)      // 4608
#define K_    2304            // kept tokens per batch
#define NUM_TOP  2074
#define NUM_RAND 230
#define DINNER 512
#define DSTATE 16
#define DCONV  4
#define DTRANK 16
#define XDBLW  (DTRANK + 2*DSTATE)   // 48
#define HID   1024
#define EPSF  1e-5f

typedef float v2f __attribute__((ext_vector_type(2)));
typedef float v8f __attribute__((ext_vector_type(8)));

__device__ __forceinline__ float siluf(float x){ return x / (1.f + expf(-x)); }
__device__ __forceinline__ float softplusf(float x){
  return x > 0.f ? x + log1pf(expf(-x)) : log1pf(expf(x));
}
__device__ __forceinline__ float geluf(float x){
  return 0.5f * x * (1.f + erff(x * 0.7071067811865475f));
}

// ---------------- permute (B,T,C,H,W) -> (B,L,C) ----------------
__global__ void permute_kernel(const float* __restrict__ xin, float* __restrict__ x){
  size_t i = (size_t)blockIdx.x*blockDim.x + threadIdx.x;
  if (i >= (size_t)B_*L_*C_) return;
  int c = (int)(i % C_); size_t r = i / C_;
  int l = (int)(r % L_); int b = (int)(r / L_);
  int t = l / (H_*W_);   int hw = l % (H_*W_);
  x[i] = xin[(((size_t)b*T_ + t)*C_ + c)*(size_t)(H_*W_) + hw];
}

// ---------------- RMSNorm (wave32 per token), optional addend + norm out ----------------
__global__ void rms_kernel(const float* __restrict__ in, const float* __restrict__ add,
                           const float* __restrict__ w, float* __restrict__ out,
                           float* __restrict__ norms, int ntok){
  int wave = threadIdx.x >> 5, lane = threadIdx.x & 31;
  int tok  = blockIdx.x * (blockDim.x >> 5) + wave;
  if (tok >= ntok) return;
  const float* p = in + (size_t)tok * C_;
  float v[8]; float ss = 0.f;
  #pragma unroll
  for (int j=0;j<8;j++){
    int c = lane + j*32;
    float t = p[c];
    if (add) t += add[(size_t)tok*C_ + c];
    v[j] = t; ss += t*t;
  }
  #pragma unroll
  for (int o=16;o>0;o>>=1) ss += __shfl_xor(ss, o, 32);
  float inv = rsqrtf(ss * (1.f/(float)C_) + EPSF);
  float s2 = 0.f;
  #pragma unroll
  for (int j=0;j<8;j++){
    int c = lane + j*32;
    float o = v[j] * inv * w[c];
    out[(size_t)tok*C_ + c] = o;
    s2 += o*o;
  }
  if (norms){
    #pragma unroll
    for (int o=16;o>0;o>>=1) s2 += __shfl_xor(s2, o, 32);
    if (lane == 0) norms[tok] = sqrtf(s2);
  }
}

// ---------------- bitonic sort on u64 keys in LDS ----------------
__device__ void bitonic_u64(unsigned long long* a, int n, bool desc, int tid, int nt){
  __syncthreads();
  for (int k=2;k<=n;k<<=1){
    for (int j=k>>1;j>0;j>>=1){
      for (int i=tid;i<n;i+=nt){
        int p = i ^ j;
        if (p > i){
          unsigned long long x = a[i], y = a[p];
          bool up = ((i & k) == 0);
          bool sw = up ? (x > y) : (x < y);
          if (desc) sw = !sw;
          if (sw){ a[i] = y; a[p] = x; }
        }
      }
      __syncthreads();
    }
  }
}

// one block per batch; dynamic LDS: 8192 u64 keys + L_ mask bytes
__global__ __launch_bounds__(1024) void select_kernel(const float* __restrict__ norms,
                                                      const float* __restrict__ rnd,
                                                      int* __restrict__ idx){
  extern __shared__ unsigned char smem[];
  unsigned long long* keys = (unsigned long long*)smem;
  unsigned char* mask = (unsigned char*)(smem + 8192*sizeof(unsigned long long));
  int b = blockIdx.x, tid = threadIdx.x, nt = blockDim.x;
  // phase 1: top-k by norm (descending, ties -> lower index first)
  for (int i=tid;i<8192;i+=nt){
    if (i < L_){
      unsigned fb = __float_as_uint(norms[(size_t)b*L_ + i]);
      keys[i] = ((unsigned long long)fb << 32) | (unsigned)(0xFFFFFFFFu - (unsigned)i);
      mask[i] = 0;
    } else keys[i] = 0ull;
  }
  bitonic_u64(keys, 8192, true, tid, nt);
  for (int j=tid;j<NUM_TOP;j+=nt){
    unsigned l = 0xFFFFFFFFu - (unsigned)keys[j];
    mask[l] = 1;
  }
  __syncthreads();
  // phase 2: random picks among the rest
  for (int i=tid;i<8192;i+=nt){
    if (i < L_ && !mask[i]){
      unsigned fb = __float_as_uint(rnd[(size_t)b*L_ + i]);
      keys[i] = ((unsigned long long)fb << 32) | (unsigned)(0xFFFFFFFFu - (unsigned)i);
    } else keys[i] = 0ull;
  }
  bitonic_u64(keys, 8192, true, tid, nt);
  for (int j=tid;j<NUM_RAND;j+=nt){
    unsigned l = 0xFFFFFFFFu - (unsigned)keys[j];
    mask[l] = 1;
  }
  __syncthreads();
  // phase 3: ascending compaction of the selected set
  for (int i=tid;i<8192;i+=nt)
    keys[i] = (i < L_ && mask[i]) ? (unsigned long long)i : 0xFFFFFFFFFFFFFFFFull;
  bitonic_u64(keys, 8192, false, tid, nt);
  for (int j=tid;j<K_;j+=nt) idx[(size_t)b*K_ + j] = (int)(unsigned)keys[j];
}

// ---------------- gather kept tokens ----------------
__global__ void gather_kernel(const float* __restrict__ xnorm, const float* __restrict__ x,
                              const int* __restrict__ idx, float* __restrict__ xs,
                              float* __restrict__ xat){
  size_t i = (size_t)blockIdx.x*blockDim.x + threadIdx.x;
  if (i >= (size_t)B_*K_*C_) return;
  int c = (int)(i % C_); size_t r = i / C_;
  int k = (int)(r % K_); int b = (int)(r / K_);
  int l = idx[b*K_ + k];
  size_t src = ((size_t)b*L_ + l)*C_ + c;
  xs[i]  = xnorm[src];
  xat[i] = x[src];
}

// ---------------- generic WMMA f32 GEMM:  C[M,N] = A[M,Kd] * W[N,Kd]^T (+bias)(+softplus) ----
// Register-blocked: each wave computes a 32x64 macro-tile (2 M-tiles x 4 N-tiles, 8 acc).
// 8 waves/block tile 512 columns. M must be a multiple of 32. Ragged N handled by
// wave-uniform tile clamping (EXEC stays all-ones for WMMA). epi: 0=none,1=bias,2=bias+softplus
__global__ __launch_bounds__(256) void wmma_gemm_kernel(
    const float* __restrict__ A, int lda,
    const float* __restrict__ Bw, int ldb,
    float* __restrict__ Cc, int ldc,
    int M, int N, int Kd,
    const float* __restrict__ bias, int epi){
  int wave = threadIdx.x >> 5, lane = threadIdx.x & 31;
  int m0 = blockIdx.x * 32;
  int nb = (blockIdx.y * 8 + wave) * 64;
  if (m0 >= M || nb >= N) return;           // uniform per wave -> EXEC all ones
  int half = lane >> 4;                     // 0: K pair {0,1}; 1: K pair {2,3}
  int lm   = lane & 15;
  const float* Ar0 = A + (size_t)(m0 + lm) * lda + half*2;
  const float* Ar1 = Ar0 + (size_t)16 * lda;
  const float* Br[4];
  bool valid[4];
  #pragma unroll
  for (int t=0;t<4;t++){
    int ntile = nb + t*16;
    valid[t] = (ntile < N);
    int ncl = valid[t] ? ntile : (N - 16);  // clamp: redundant compute, store guarded
    Br[t] = Bw + (size_t)(ncl + lm) * ldb + half*2;
  }
  v8f acc0[4], acc1[4];
  const v8f vzero = {0.f,0.f,0.f,0.f,0.f,0.f,0.f,0.f};
  #pragma unroll
  for (int t=0;t<4;t++){ acc0[t] = vzero; acc1[t] = vzero; }
  for (int k=0;k<Kd;k+=4){
    v2f a0 = *(const v2f*)(Ar0 + k);
    v2f a1 = *(const v2f*)(Ar1 + k);
    #pragma unroll
    for (int t=0;t<4;t++){
      v2f b = *(const v2f*)(Br[t] + k);
      acc0[t] = __builtin_amdgcn_wmma_f32_16x16x4_f32(false, a0, false, b,
                                                      (short)0, acc0[t], false, false);
      acc1[t] = __builtin_amdgcn_wmma_f32_16x16x4_f32(false, a1, false, b,
                                                      (short)0, acc1[t], false, false);
    }
  }
  int mb0 = m0 + half*8;
  #pragma unroll
  for (int t=0;t<4;t++){
    if (!valid[t]) continue;                // wave-uniform
    int nc = nb + t*16 + lm;
    float bv = (epi >= 1 && bias) ? bias[nc] : 0.f;
    #pragma unroll
    for (int r=0;r<8;r++){
      float v0 = acc0[t][r] + bv;
      float v1 = acc1[t][r] + bv;
      if (epi == 2){ v0 = softplusf(v0); v1 = softplusf(v1); }
      Cc[(size_t)(mb0 + r)*ldc + nc]      = v0;
      Cc[(size_t)(mb0 + 16 + r)*ldc + nc] = v1;
    }
  }
}

// ---------------- direction-aware causal conv1d (+bias+silu) ----------------
// uz: (B,K,1024) rows in original order; udir: ((dir*B+b)*K+t, 512) in direction order
__global__ void conv1d_kernel(const float* __restrict__ uz, const float* __restrict__ cw,
                              const float* __restrict__ cb, float* __restrict__ udir){
  size_t i = (size_t)blockIdx.x*blockDim.x + threadIdx.x;
  if (i >= (size_t)2*B_*K_*DINNER) return;
  int d = (int)(i & (DINNER-1)); size_t r = i >> 9;
  int t = (int)(r % K_); r /= K_;
  int b = (int)(r % B_); int dir = (int)(r / B_);
  float acc = cb[d];
  #pragma unroll
  for (int j=0;j<DCONV;j++){
    int p = t - (DCONV-1) + j;
    if (p >= 0){
      int pos = dir ? (K_-1-p) : p;
      acc += cw[d*DCONV + j] * uz[((size_t)b*K_ + pos)*(size_t)(2*DINNER) + d];
    }
  }
  udir[i] = siluf(acc);
}

// ---------------- selective scan: one block per (batch, direction), lane = channel ----------
#define SCHUNK 32
__global__ __launch_bounds__(512) void scan_kernel(
    const float* __restrict__ xdbl, const float* __restrict__ delta,
    const float* __restrict__ udir, const float* __restrict__ uz,
    const float* __restrict__ A_log, const float* __restrict__ Dv,
    float* __restrict__ y2){
  __shared__ float sBC[SCHUNK][32];     // cols 0..15 = B_t, 16..31 = C_t
  int b = blockIdx.x, dir = blockIdx.y;
  int d = threadIdx.x;
  float Arow[DSTATE];
  #pragma unroll
  for (int s=0;s<DSTATE;s++) Arow[s] = -expf(A_log[d*DSTATE + s]);
  float Dd = Dv[d];
  float h[DSTATE];
  #pragma unroll
  for (int s=0;s<DSTATE;s++) h[s] = 0.f;
  size_t rowbase = ((size_t)dir*B_ + b) * K_;
  for (int t0=0; t0<K_; t0+=SCHUNK){
    __syncthreads();
    for (int e=threadIdx.x; e<SCHUNK*32; e+=512){
      int rr = e >> 5, cc = e & 31;
      sBC[rr][cc] = xdbl[(rowbase + t0 + rr)*XDBLW + DTRANK + cc];
    }
    __syncthreads();
    for (int tl=0; tl<SCHUNK; tl++){
      int t = t0 + tl;
      size_t row = rowbase + t;
      float dlt = delta[row*DINNER + d];
      float uu  = udir [row*DINNER + d];
      float du  = dlt * uu;
      float y = 0.f;
      #pragma unroll
      for (int s=0;s<DSTATE;s++){
        float e = expf(dlt * Arow[s]);
        h[s] = e*h[s] + du * sBC[tl][s];
        y += h[s] * sBC[tl][DSTATE + s];
      }
      y += uu * Dd;
      int gpos = dir ? (K_-1-t) : t;
      float zz = uz[((size_t)b*K_ + gpos)*(size_t)(2*DINNER) + DINNER + d];
      y2[row*DINNER + d] = y * siluf(zz);
    }
  }
}

// ---------------- combine forward + flipped-reverse before out-projection ----------------
__global__ void y2comb_kernel(const float* __restrict__ y2d, float* __restrict__ y2c){
  size_t i = (size_t)blockIdx.x*blockDim.x + threadIdx.x;
  if (i >= (size_t)B_*K_*DINNER) return;
  int d = (int)(i & (DINNER-1)); size_t r = i >> 9;
  int t = (int)(r % K_); int b = (int)(r / K_);
  size_t f  = ((size_t)b*K_ + t)*DINNER + d;
  size_t rv = (((size_t)B_ + b)*K_ + (K_-1-t))*DINNER + d;
  y2c[i] = y2d[f] + y2d[rv];
}

// ---------------- scatter fc1 output into dense (B,L,HID) ----------------
__global__ void scatter_kernel(const float* __restrict__ h1, const int* __restrict__ idx,
                               float* __restrict__ dense){
  size_t i = (size_t)blockIdx.x*blockDim.x + threadIdx.x;
  if (i >= (size_t)B_*K_*HID) return;
  int c = (int)(i & (HID-1)); size_t r = i >> 10;
  int k = (int)(r % K_); int b = (int)(r / K_);
  int l = idx[b*K_ + k];
  dense[((size_t)b*L_ + l)*HID + c] = h1[i];
}

// ---------------- depthwise 3x3 over (H,W) + gather kept + exact GELU ----------------
__global__ void dwconv_kernel(const float* __restrict__ dense, const int* __restrict__ idx,
                              const float* __restrict__ dw, const float* __restrict__ db,
                              float* __restrict__ hs){
  size_t i = (size_t)blockIdx.x*blockDim.x + threadIdx.x;
  if (i >= (size_t)B_*K_*HID) return;
  int c = (int)(i & (HID-1)); size_t r = i >> 10;
  int k = (int)(r % K_); int b = (int)(r / K_);
  int l = idx[b*K_ + k];
  int t = l / (H_*W_); int hw = l % (H_*W_);
  int hy = hw / W_;    int wx = hw % W_;
  float acc = db[c];
  #pragma unroll
  for (int dy=0;dy<3;dy++){
    int yy = hy + dy - 1;
    if (yy < 0 || yy >= H_) continue;
    #pragma unroll
    for (int dx=0;dx<3;dx++){
      int xx = wx + dx - 1;
      if (xx < 0 || xx >= W_) continue;
      acc += dw[c*9 + dy*3 + dx] *
             dense[((size_t)b*L_ + ((size_t)t*(H_*W_) + yy*W_ + xx))*HID + c];
    }
  }
  hs[i] = geluf(acc);
}

// ---------------- final scatter-add residual update into (B,T,C,H,W) output ----------------
__global__ void finaladd_kernel(const float* __restrict__ mo, const float* __restrict__ mlo,
                                const int* __restrict__ idx, float* __restrict__ out){
  size_t i = (size_t)blockIdx.x*blockDim.x + threadIdx.x;
  if (i >= (size_t)B_*K_*C_) return;
  int c = (int)(i & (C_-1)); size_t r = i >> 8;
  int k = (int)(r % K_); int b = (int)(r / K_);
  int l = idx[b*K_ + k];
  int t = l / (H_*W_); int hw = l % (H_*W_);
  out[(((size_t)b*T_ + t)*C_ + c)*(size_t)(H_*W_) + hw] += mo[i] + mlo[i];
}

// =======================================================================================
extern "C" void kernel_launch(void* const* d_in, const int* in_sizes, int n_in,
                              void* d_out, int out_size, void* d_ws, size_t ws_size,
                              hipStream_t stream) {
  (void)in_sizes; (void)n_in; (void)out_size; (void)ws_size;
  const float* x_in    = (const float*)d_in[0];
  const float* rnd     = (const float*)d_in[1];
  const float* n1w     = (const float*)d_in[2];
  const float* n2w     = (const float*)d_in[3];
  const float* W_in    = (const float*)d_in[4];
  const float* conv_w  = (const float*)d_in[5];
  const float* conv_b  = (const float*)d_in[6];
  const float* W_xproj = (const float*)d_in[7];
  const float* W_dt    = (const float*)d_in[8];
  const float* b_dt    = (const float*)d_in[9];
  const float* A_log   = (const float*)d_in[10];
  const float* Dv      = (const float*)d_in[11];
  const float* W_out   = (const float*)d_in[12];
  const float* fc1_w   = (const float*)d_in[13];
  const float* fc1_b   = (const float*)d_in[14];
  const float* dw_w    = (const float*)d_in[15];
  const float* dw_b    = (const float*)d_in[16];
  const float* fc2_w   = (const float*)d_in[17];
  const float* fc2_b   = (const float*)d_in[18];
  float* out = (float*)d_out;

  char* ws = (char*)d_ws;
  size_t off = 0;
  auto alloc = [&](size_t nfl)->float* {
    float* p = (float*)(ws + off);
    off += ((nfl*sizeof(float) + 255) & ~(size_t)255);
    return p;
  };
  float* x     = alloc((size_t)B_*L_*C_);
  float* xnorm = alloc((size_t)B_*L_*C_);
  float* norms = alloc((size_t)B_*L_);
  int*   idx   = (int*)alloc((size_t)B_*K_);
  float* xs    = alloc((size_t)B_*K_*C_);
  float* xat   = alloc((size_t)B_*K_*C_);
  float* uzb   = alloc((size_t)B_*K_*2*DINNER);
  float* udir  = alloc((size_t)2*B_*K_*DINNER);
  float* xdbl  = alloc((size_t)2*B_*K_*XDBLW);
  float* delta = alloc((size_t)2*B_*K_*DINNER);
  float* y2d   = alloc((size_t)2*B_*K_*DINNER);
  float* y2c   = alloc((size_t)B_*K_*DINNER);
  float* mo    = alloc((size_t)B_*K_*C_);
  float* xs2   = alloc((size_t)B_*K_*C_);
  float* h1    = alloc((size_t)B_*K_*HID);
  float* dense = alloc((size_t)B_*L_*HID);
  float* hs    = alloc((size_t)B_*K_*HID);
  float* mlo   = alloc((size_t)B_*K_*C_);

  auto cdiv = [](size_t a, size_t b)->unsigned { return (unsigned)((a + b - 1)/b); };
  const int TB = 256;
  const int Mt = B_*K_;        // 4608 token rows
  const int Md = 2*B_*K_;      // 9216 direction-stacked rows

  // 1) permute + RMSNorm1 + token norms
  permute_kernel<<<cdiv((size_t)B_*L_*C_,TB), TB, 0, stream>>>(x_in, x);
  rms_kernel<<<cdiv((size_t)B_*L_,8), TB, 0, stream>>>(x, nullptr, n1w, xnorm, norms, B_*L_);

  // 2) sparsify: top-k + random + ascending index sort (one block per batch)
  size_t smem = 8192*sizeof(unsigned long long) + L_;
  select_kernel<<<B_, 1024, smem, stream>>>(norms, rnd, idx);

  // 3) gather kept tokens
  gather_kernel<<<cdiv((size_t)B_*K_*C_,TB), TB, 0, stream>>>(xnorm, x, idx, xs, xat);

  // 4) mamba in-projection: uz = xs @ W_in^T   (M=4608, N=1024, K=256)
  wmma_gemm_kernel<<<dim3(Mt/32, cdiv(2*DINNER,512)), TB, 0, stream>>>(
      xs, C_, W_in, C_, uzb, 2*DINNER, Mt, 2*DINNER, C_, nullptr, 0);

  // 5) causal conv1d + silu, both directions
  conv1d_kernel<<<cdiv((size_t)2*B_*K_*DINNER,TB), TB, 0, stream>>>(uzb, conv_w, conv_b, udir);

  // 6) x-projection: xdbl = u_dir @ W_xproj^T  (M=9216, N=48, K=512)
  wmma_gemm_kernel<<<dim3(Md/32, 1), TB, 0, stream>>>(
      udir, DINNER, W_xproj, DINNER, xdbl, XDBLW, Md, XDBLW, DINNER, nullptr, 0);

  // 7) delta = softplus(dt @ W_dt^T + b_dt)    (M=9216, N=512, K=16, lda=48)
  wmma_gemm_kernel<<<dim3(Md/32, 1), TB, 0, stream>>>(
      xdbl, XDBLW, W_dt, DTRANK, delta, DINNER, Md, DINNER, DTRANK, b_dt, 2);

  // 8) selective scan (batch x direction), fused +u*D and *silu(z)
  scan_kernel<<<dim3(B_, 2), 512, 0, stream>>>(xdbl, delta, udir, uzb, A_log, Dv, y2d);

  // 9) combine directions, then out-projection (M=4608, N=256, K=512)
  y2comb_kernel<<<cdiv((size_t)B_*K_*DINNER,TB), TB, 0, stream>>>(y2d, y2c);
  wmma_gemm_kernel<<<dim3(Mt/32, 1), TB, 0, stream>>>(
      y2c, DINNER, W_out, DINNER, mo, C_, Mt, C_, DINNER, nullptr, 0);

  // 10) mid = x_at + mamba_out ; RMSNorm2
  rms_kernel<<<cdiv((size_t)B_*K_,8), TB, 0, stream>>>(xat, mo, n2w, xs2, nullptr, B_*K_);

  // 11) fc1 (M=4608, N=1024, K=256) + bias
  wmma_gemm_kernel<<<dim3(Mt/32, cdiv(HID,512)), TB, 0, stream>>>(
      xs2, C_, fc1_w, C_, h1, HID, Mt, HID, C_, fc1_b, 1);

  // 12) scatter into dense grid, depthwise 3x3, gather + GELU
  hipMemsetAsync(dense, 0, (size_t)B_*L_*HID*sizeof(float), stream);
  scatter_kernel<<<cdiv((size_t)B_*K_*HID,TB), TB, 0, stream>>>(h1, idx, dense);
  dwconv_kernel<<<cdiv((size_t)B_*K_*HID,TB), TB, 0, stream>>>(dense, idx, dw_w, dw_b, hs);

  // 13) fc2 (M=4608, N=256, K=1024) + bias
  wmma_gemm_kernel<<<dim3(Mt/32, 1), TB, 0, stream>>>(
      hs, HID, fc2_w, HID, mlo, C_, Mt, C_, HID, fc2_b, 1);

  // 14) out = x_in (shortcut in original layout) + scatter(update)
  hipMemcpyAsync(out, x_in, (size_t)B_*T_*C_*H_*W_*sizeof(float),
                 hipMemcpyDeviceToDevice, stream);
  finaladd_kernel<<<cdiv((size_t)B_*K_*C_,TB), TB, 0, stream>>>(mo, mlo, idx, out);
}